// CorrespondenceLoss_73735998537891
// MI455X (gfx1250) — compile-verified
//
#include <hip/hip_runtime.h>

// ---------------------------------------------------------------------------
// CorrespondenceLoss for MI455X (gfx1250, wave32).
// Gather-bound kernel (~25MB gathered @ 23.3TB/s => ~1.1us floor).
// Uses V_WMMA_F32_16X16X4_F32 (full-precision FP32 WMMA): one wave handles 16
// keypoints, computing diag(S*T^T), diag(S*S^T), diag(T*T^T) = dot / |s|^2 /
// |t|^2 via three WMMAs per K-chunk of 4.
// ---------------------------------------------------------------------------

typedef float v2f __attribute__((ext_vector_type(2)));
typedef float v8f __attribute__((ext_vector_type(8)));

#define B_   16
#define H_   64
#define W_   64
#define D_   768
#define N_   256
#define NWAVES_ (B_ * N_ / 16)   // 256 waves total
#define EPS_ 1e-8f

__device__ __forceinline__ float sel8(v8f a, int v) {
    // branchless 8-way select (v_cndmask chain)
    float r = a[0];
    r = (v == 1) ? a[1] : r;
    r = (v == 2) ? a[2] : r;
    r = (v == 3) ? a[3] : r;
    r = (v == 4) ? a[4] : r;
    r = (v == 5) ? a[5] : r;
    r = (v == 6) ? a[6] : r;
    r = (v == 7) ? a[7] : r;
    return r;
}

__global__ __launch_bounds__(256) void corr_loss_wmma_kernel(
    const float* __restrict__ src, const float* __restrict__ tgt,
    const int* __restrict__ skps, const int* __restrict__ tkps,
    const unsigned char* __restrict__ mask, const int* __restrict__ patch_p,
    float* __restrict__ partials)
{
    const int patch  = *patch_p;                       // scalar (smem load)
    const int lane   = threadIdx.x & 31;
    const int waveId = blockIdx.x * (blockDim.x >> 5) + (threadIdx.x >> 5);

    // Fragment row owned by this lane: lanes L and L+16 both serve row L&15.
    const int r  = lane & 15;
    const int kp = waveId * 16 + r;                    // flattened (b,n)
    const int b  = kp >> 8;                            // N_ == 256
    const int n  = kp & (N_ - 1);

    // keypoint -> clamped patch coords -> gathered row base pointers
    int sxi = skps[(b * N_ + n) * 2 + 0] / patch;
    int syi = skps[(b * N_ + n) * 2 + 1] / patch;
    int txi = tkps[(b * N_ + n) * 2 + 0] / patch;
    int tyi = tkps[(b * N_ + n) * 2 + 1] / patch;
    sxi = min(max(sxi, 0), W_ - 1);  syi = min(max(syi, 0), H_ - 1);
    txi = min(max(txi, 0), W_ - 1);  tyi = min(max(tyi, 0), H_ - 1);

    const float* __restrict__ srow = src + (((size_t)b * H_ + syi) * W_ + sxi) * D_;
    const float* __restrict__ trow = tgt + (((size_t)b * H_ + tyi) * W_ + txi) * D_;

    // 16x4 f32 A fragment layout == 4x16 f32 B fragment layout per lane:
    // lanes 0-15 hold {k, k+1} of row L; lanes 16-31 hold {k+2, k+3}.
    const int kofs = (lane >> 4) * 2;

    v8f acc_st = {};   // diag -> dot(src_i, tgt_i)
    v8f acc_ss = {};   // diag -> |src_i|^2
    v8f acc_tt = {};   // diag -> |tgt_i|^2

    #pragma unroll 4
    for (int k = 0; k < D_; k += 4) {
        v2f a = *(const v2f*)(srow + k + kofs);        // global_load_b64
        v2f t = *(const v2f*)(trow + k + kofs);
        // v_wmma_f32_16x16x4_f32 : D = A x B + C  (full fp32)
        acc_st = __builtin_amdgcn_wmma_f32_16x16x4_f32(
            false, a, false, t, (short)0, acc_st, false, false);
        acc_ss = __builtin_amdgcn_wmma_f32_16x16x4_f32(
            false, a, false, a, (short)0, acc_ss, false, false);
        acc_tt = __builtin_amdgcn_wmma_f32_16x16x4_f32(
            false, t, false, t, (short)0, acc_tt, false, false);
    }

    // 16x16 f32 C/D layout: (i,i) at lane i / vgpr i (i<8), lane i+16 / vgpr
    // i-8 (i>=8). So lanes {0..7, 24..31} hold diag slot (lane&7).
    const int  v       = lane & 7;
    const bool hasDiag = (lane < 8) || (lane >= 24);
    const float dot = sel8(acc_st, v);
    const float ss  = sel8(acc_ss, v);
    const float tt  = sel8(acc_tt, v);

    float contrib = 0.0f, cnt = 0.0f;
    if (hasDiag) {
        const int i   = (lane & 7) + ((lane >= 16) ? 8 : 0);
        const int kpi = waveId * 16 + i;
        const int bb  = kpi >> 8;
        const int nn  = kpi & (N_ - 1);
        const float valid = mask[bb * N_ + nn] ? 1.0f : 0.0f;
        const float denom = fmaxf(sqrtf(ss) * sqrtf(tt), EPS_);
        const float cosv  = dot / denom;
        contrib = (1.0f - cosv) * valid;
        cnt     = valid;
    }

    // wave32 butterfly reduction (ds_swizzle/permlane)
    #pragma unroll
    for (int o = 16; o >= 1; o >>= 1) {
        contrib += __shfl_xor(contrib, o, 32);
        cnt     += __shfl_xor(cnt,     o, 32);
    }
    if (lane == 0) {
        partials[waveId * 2 + 0] = contrib;
        partials[waveId * 2 + 1] = cnt;
    }
}

// Deterministic fixed-order tree reduction of the 256 per-wave partials.
__global__ __launch_bounds__(NWAVES_) void corr_loss_finish_kernel(
    const float* __restrict__ partials, float* __restrict__ out)
{
    __shared__ float sT[NWAVES_];
    __shared__ float sC[NWAVES_];
    const int t = threadIdx.x;
    sT[t] = partials[t * 2 + 0];
    sC[t] = partials[t * 2 + 1];
    __syncthreads();
    #pragma unroll
    for (int s = NWAVES_ / 2; s >= 1; s >>= 1) {
        if (t < s) { sT[t] += sT[t + s]; sC[t] += sC[t + s]; }
        __syncthreads();
    }
    if (t == 0) out[0] = sT[0] / fmaxf(sC[0], 1.0f);
}

extern "C" void kernel_launch(void* const* d_in, const int* in_sizes, int n_in,
                              void* d_out, int out_size, void* d_ws, size_t ws_size,
                              hipStream_t stream) {
    const float*         src   = (const float*)d_in[0];
    const float*         tgt   = (const float*)d_in[1];
    const int*           skps  = (const int*)d_in[2];
    const int*           tkps  = (const int*)d_in[3];
    const unsigned char* mask  = (const unsigned char*)d_in[4];  // jax bool = 1 byte
    const int*           patch = (const int*)d_in[5];
    float*               out   = (float*)d_out;
    float*               parts = (float*)d_ws;                   // 256*2 floats

    // 256 waves total, 8 waves (256 threads) per block -> 32 blocks
    corr_loss_wmma_kernel<<<NWAVES_ / 8, 256, 0, stream>>>(
        src, tgt, skps, tkps, mask, patch, parts);
    corr_loss_finish_kernel<<<1, NWAVES_, 0, stream>>>(parts, out);
}